// Model_69767448756493
// MI455X (gfx1250) — compile-verified
//
#include <hip/hip_runtime.h>
#include <hip/hip_bf16.h>
#include <stdint.h>

// out[output_offset[t] + i] = input_offset[permute[t]] + i   for i in [0, len_t)
//
// One table per blockIdx.y; blockIdx.x splits the table's ramp across
// CHUNKS blocks. Uniform (SGPR) loads of the per-table descriptors,
// then pure streaming b128 stores of a register-synthesized ramp.

#ifndef RAMP_BLOCK
#define RAMP_BLOCK 256   // 8 wave32 waves per workgroup
#endif
#ifndef RAMP_CHUNKS
#define RAMP_CHUNKS 16   // blocks per table
#endif

__global__ __launch_bounds__(RAMP_BLOCK) void table_ramp_kernel(
    const int* __restrict__ permute,
    const int* __restrict__ input_offset,
    const int* __restrict__ output_offset,
    int* __restrict__ out) {
  const int t = blockIdx.y;

  // All four loads are uniform across the block -> s_load via constant cache.
  const int o_beg = output_offset[t];
  const int o_end = output_offset[t + 1];
  const int len   = o_end - o_beg;
  const int base  = input_offset[permute[t]];

  int* __restrict__ dst = out + o_beg;

  const int nthreads = blockDim.x * gridDim.x;                 // threads covering this table
  const int tid      = blockIdx.x * blockDim.x + threadIdx.x;

  // --- head: scalar stores until dst is 16-byte aligned ---
  const uintptr_t addr = (uintptr_t)dst;
  int head = (int)(((16u - (unsigned)(addr & 15u)) & 15u) >> 2);
  if (head > len) head = len;
  if (tid < head) dst[tid] = base + tid;                       // head < 4, one pass

  // --- body: int4 ramp -> global_store_b128 ---
  const int body = (len - head) >> 2;                          // number of int4 chunks
  int4* __restrict__ dst4 = (int4*)(dst + head);
  for (int v = tid; v < body; v += nthreads) {
    const int i = base + head + (v << 2);
    int4 val;
    val.x = i;
    val.y = i + 1;
    val.z = i + 2;
    val.w = i + 3;
    dst4[v] = val;
  }

  // --- tail: remaining 0..3 elements ---
  const int done = head + (body << 2);
  const int rem  = len - done;                                 // < 4
  if (tid < rem) dst[done + tid] = base + done + tid;
}

extern "C" void kernel_launch(void* const* d_in, const int* in_sizes, int n_in,
                              void* d_out, int out_size, void* d_ws, size_t ws_size,
                              hipStream_t stream) {
  (void)n_in; (void)out_size; (void)d_ws; (void)ws_size;

  const int* permute       = (const int*)d_in[0];  // [tables]
  const int* input_offset  = (const int*)d_in[1];  // [tables+1]
  const int* output_offset = (const int*)d_in[2];  // [tables+1]
  // d_in[3] is output_size (scalar); out_size carries the same info.

  int* out = (int*)d_out;

  const int tables = in_sizes[0];

  dim3 grid(RAMP_CHUNKS, tables, 1);
  dim3 block(RAMP_BLOCK, 1, 1);
  table_ramp_kernel<<<grid, block, 0, stream>>>(permute, input_offset, output_offset, out);
}